// BoWDecoder_46170898432277
// MI455X (gfx1250) — compile-verified
//
#include <hip/hip_runtime.h>
#include <hip/hip_bf16.h>
#include <math.h>

// ---------------------------------------------------------------------------
// BoW decoder loss for MI455X (gfx1250, wave32).
// Memory-bound on streaming W1 (206 MB fp32) -> use native f32 WMMA
// (V_WMMA_F32_16X16X4_F32), read W1 exactly once, keep h/logits in L2.
// ---------------------------------------------------------------------------

typedef __attribute__((ext_vector_type(2))) float v2f;
typedef __attribute__((ext_vector_type(8))) float v8f;

#define BATCH   32
#define DM      1024
#define VOCAB   50257
#define SEQ     128
#define NTILE2  ((VOCAB + 15) / 16)   // 3142 column tiles for GEMM2

// ----------------------------- Kernel 1 ------------------------------------
// h = gelu(z @ W0 + b0), z:[32,1024], W0:[1024,1024] row-major [in,out].
// One wave computes one 16x16 output tile; K-loop in steps of 4 with f32 WMMA.
__global__ __launch_bounds__(32)
void k1_gemm_gelu(const float* __restrict__ z, const float* __restrict__ W0,
                  const float* __restrict__ b0, float* __restrict__ h)
{
    const int lane  = threadIdx.x;       // 0..31
    const int ntile = blockIdx.x;        // 0..63  (N = 1024 / 16)
    const int mtile = blockIdx.y;        // 0..1   (M = 32 / 16)
    const int half  = lane >> 4;         // half-wave select
    const int l15   = lane & 15;
    const int arow  = mtile * 16 + l15;  // A row for this lane
    const int col   = ntile * 16 + l15;  // B/C/D column for this lane

    v8f acc = {};
    for (int k = 0; k < DM; k += 4) {
        // A 16x4 f32: lanes 0-15 hold K=k..k+1, lanes 16-31 hold K=k+2..k+3
        v2f a = *(const v2f*)(z + (size_t)arow * DM + k + half * 2);
        // B 4x16 f32: VGPR0 = K rows {k, k+2}, VGPR1 = {k+1, k+3} by half-wave
        v2f b;
        b.x = W0[(size_t)(k + half * 2    ) * DM + col];
        b.y = W0[(size_t)(k + half * 2 + 1) * DM + col];
        acc = __builtin_amdgcn_wmma_f32_16x16x4_f32(
            false, a, false, b, (short)0, acc, false, false);
    }

    const float bias = b0[col];
#pragma unroll
    for (int i = 0; i < 8; ++i) {
        float x = acc[i] + bias;
        // exact GELU: x * 0.5 * (1 + erf(x / sqrt(2)))
        float g = 0.5f * x * (1.0f + erff(x * 0.70710678118654752440f));
        int row = mtile * 16 + half * 8 + i;
        h[(size_t)row * DM + col] = g;
    }
}

// ----------------------------- Kernel 2 ------------------------------------
// logits = h @ W1 + b1, h:[32,1024], W1:[1024,50257] row-major [in,out].
// One wave per 16-column vocab tile; two accumulators cover M=0..31.
// W1 streamed exactly once (the bandwidth floor of the whole problem).
__global__ __launch_bounds__(32)
void k2_gemm_logits(const float* __restrict__ h, const float* __restrict__ W1,
                    const float* __restrict__ b1, float* __restrict__ logits)
{
    const int lane = threadIdx.x;
    const int tile = blockIdx.x;             // 0..NTILE2-1
    const int half = lane >> 4;
    const int l15  = lane & 15;
    const int n    = tile * 16 + l15;
    const int nld  = (n < VOCAB) ? n : (VOCAB - 1);  // clamp ragged last tile

    const float* __restrict__ A0 = h + (size_t)l15 * DM;        // rows 0..15
    const float* __restrict__ A1 = h + (size_t)(16 + l15) * DM; // rows 16..31

    v8f acc0 = {};
    v8f acc1 = {};
    for (int k = 0; k < DM; k += 4) {
        v2f a0 = *(const v2f*)(A0 + k + half * 2);
        v2f a1 = *(const v2f*)(A1 + k + half * 2);
        v2f b;
        b.x = W1[(size_t)(k + half * 2    ) * VOCAB + nld];
        b.y = W1[(size_t)(k + half * 2 + 1) * VOCAB + nld];
        acc0 = __builtin_amdgcn_wmma_f32_16x16x4_f32(
            false, a0, false, b, (short)0, acc0, false, false);
        acc1 = __builtin_amdgcn_wmma_f32_16x16x4_f32(
            false, a1, false, b, (short)0, acc1, false, false);
    }

    if (n < VOCAB) {
        const float bias = b1[n];
#pragma unroll
        for (int i = 0; i < 8; ++i) {
            int r0 = half * 8 + i;          // rows 0..15
            int r1 = 16 + r0;               // rows 16..31
            logits[(size_t)r0 * VOCAB + n] = acc0[i] + bias;
            logits[(size_t)r1 * VOCAB + n] = acc1[i] + bias;
        }
    }
}

// ----------------------------- Kernel 3 ------------------------------------
// Per batch row: LSE over vocab + gather of 128 label logits.
// rowout[b] = SEQ * LSE_b - sum_s logits[b][label[b,s]]   (deterministic)
__global__ __launch_bounds__(256)
void k3_row_reduce(const float* __restrict__ logits,
                   const int* __restrict__ labels,
                   float* __restrict__ rowout)
{
    __shared__ float sdata[256];
    const int b   = blockIdx.x;
    const int tid = threadIdx.x;
    const float* __restrict__ row = logits + (size_t)b * VOCAB;

    // 1) row max
    float m = -INFINITY;
    for (int j = tid; j < VOCAB; j += 256) m = fmaxf(m, row[j]);
    sdata[tid] = m; __syncthreads();
    for (int s = 128; s > 0; s >>= 1) {
        if (tid < s) sdata[tid] = fmaxf(sdata[tid], sdata[tid + s]);
        __syncthreads();
    }
    const float rowmax = sdata[0];
    __syncthreads();

    // 2) sum of exp
    float sum = 0.0f;
    for (int j = tid; j < VOCAB; j += 256) sum += expf(row[j] - rowmax);
    sdata[tid] = sum; __syncthreads();
    for (int s = 128; s > 0; s >>= 1) {
        if (tid < s) sdata[tid] += sdata[tid + s];
        __syncthreads();
    }
    const float lse = rowmax + logf(sdata[0]);
    __syncthreads();

    // 3) label gather (SEQ = 128 labels per row)
    float g = 0.0f;
    if (tid < SEQ) {
        int idx = labels[b * SEQ + tid];
        g = row[idx];
    }
    sdata[tid] = g; __syncthreads();
    for (int s = 128; s > 0; s >>= 1) {
        if (tid < s) sdata[tid] += sdata[tid + s];
        __syncthreads();
    }
    if (tid == 0) rowout[b] = (float)SEQ * lse - sdata[0];
}

// ----------------------------- Kernel 4 ------------------------------------
// One wave32 sums the 32 per-row partials -> scalar mean.
__global__ __launch_bounds__(32)
void k4_final(const float* __restrict__ rowout, float* __restrict__ out)
{
    int lane = threadIdx.x;
    float v = rowout[lane];
#pragma unroll
    for (int off = 16; off > 0; off >>= 1) v += __shfl_down(v, off, 32);
    if (lane == 0) out[0] = v / (float)(BATCH * SEQ);
}

// ---------------------------------------------------------------------------
extern "C" void kernel_launch(void* const* d_in, const int* in_sizes, int n_in,
                              void* d_out, int out_size, void* d_ws, size_t ws_size,
                              hipStream_t stream)
{
    // setup_inputs() order: z, labels, W0, b0, W1, b1
    const float* z      = (const float*)d_in[0];
    const int*   labels = (const int*)  d_in[1];
    const float* W0     = (const float*)d_in[2];
    const float* b0     = (const float*)d_in[3];
    const float* W1     = (const float*)d_in[4];
    const float* b1     = (const float*)d_in[5];
    float* out = (float*)d_out;

    // workspace layout
    char* ws = (char*)d_ws;
    float* h      = (float*)ws;                                   // 32*1024*4   = 128 KB
    float* logits = (float*)(ws + (size_t)BATCH * DM * 4);        // 32*50257*4  ~ 6.43 MB
    float* rowout = (float*)(ws + (size_t)BATCH * DM * 4
                                + (size_t)BATCH * VOCAB * 4);     // 32*4 B

    k1_gemm_gelu  <<<dim3(DM / 16, BATCH / 16), 32, 0, stream>>>(z, W0, b0, h);
    k2_gemm_logits<<<NTILE2,                    32, 0, stream>>>(h, W1, b1, logits);
    k3_row_reduce <<<BATCH,                    256, 0, stream>>>(logits, labels, rowout);
    k4_final      <<<1,                         32, 0, stream>>>(rowout, out);
}